// HashCounter_64493228917034
// MI455X (gfx1250) — compile-verified
//
#include <hip/hip_runtime.h>

// Reference shapes (fixed by setup_inputs).
#define DIM         64
#define CODE_LEN    16
#define NUM_BUCKETS 65536

typedef float v2f __attribute__((ext_vector_type(2)));
typedef float v8f __attribute__((ext_vector_type(8)));

// Copy initial counts into the counts region of d_out (d_out is poisoned).
__global__ void hc_init_counts(const float* __restrict__ counts_in,
                               float* __restrict__ out_counts) {
    int i = blockIdx.x * blockDim.x + threadIdx.x;
    if (i < NUM_BUCKETS) out_counts[i] = counts_in[i];
}

// One wave32 computes a 16-row tile: scores = items[16x64] @ W[64x16] via
// 16 chained V_WMMA_F32_16X16X4_F32, then sign-bits -> bucket id -> outputs.
__global__ __launch_bounds__(256) void hc_hash_count(
    const float* __restrict__ items,
    const float* __restrict__ W,        // [64 x 16], row-major
    float* __restrict__ out_buckets,    // [N] bucket ids (stored as float)
    float* __restrict__ out_counts,     // [NUM_BUCKETS]
    int n_items)
{
    const unsigned tid   = blockIdx.x * blockDim.x + threadIdx.x;
    const unsigned wave  = tid >> 5;
    const unsigned lane  = threadIdx.x & 31u;
    const unsigned row0  = wave * 16u;
    if (row0 + 16u > (unsigned)n_items) return;   // wave-uniform guard

    const unsigned rloc  = lane & 15u;   // A: row / B,C,D: column
    const unsigned khalf = lane >> 4;    // 0: K pair {0,1}; 1: K pair {2,3}

    // A fragment source: items[row0+rloc][4k + 2*khalf + {0,1}]
    const float* itemRow = items + (size_t)(row0 + rloc) * DIM + 2u * khalf;
    // B fragment source: W[4k + 2*khalf + {0,1}][rloc]
    const float* wBase   = W + rloc + (2u * khalf) * CODE_LEN;

    v8f c = {0.f, 0.f, 0.f, 0.f, 0.f, 0.f, 0.f, 0.f};

    #pragma unroll
    for (int k = 0; k < DIM / 4; ++k) {
        v2f a = *(const v2f*)(itemRow + 4 * k);        // 8B aligned
        v2f b;
        b.x = wBase[(4 * k + 0) * CODE_LEN];
        b.y = wBase[(4 * k + 1) * CODE_LEN];
        // D = A(16x4,f32) * B(4x16,f32) + C ; 8-arg f32 WMMA form
        c = __builtin_amdgcn_wmma_f32_16x16x4_f32(
                /*neg_a=*/false, a, /*neg_b=*/false, b,
                /*c_mod=*/(short)0, c, /*reuse_a=*/false, /*reuse_b=*/false);
    }

    // Lane holds column `rloc` of rows (row0 + 8*khalf + 0..7) in c[0..7].
    // Sign bit for column n contributes 2^(15-n). Pack 2 rows per word and
    // OR-reduce across the 16-lane half-group (xor masks stay in-group).
    const unsigned bit = 1u << (15u - rloc);
    unsigned codes[4];
    #pragma unroll
    for (int p = 0; p < 4; ++p) {
        unsigned w = ((c[2 * p]     >= 0.f) ? bit : 0u)
                   | ((c[2 * p + 1] >= 0.f) ? (bit << 16) : 0u);
        #pragma unroll
        for (int m = 1; m <= 8; m <<= 1)
            w |= (unsigned)__shfl_xor((int)w, m, 32);
        codes[p] = w;
    }

    // Lanes 0..7 of each half-group emit one row each (16 rows / wave).
    if (rloc < 8u) {
        unsigned w      = codes[rloc >> 1];
        unsigned bucket = (rloc & 1u) ? (w >> 16) : (w & 0xFFFFu);
        unsigned row    = row0 + 8u * khalf + rloc;
        out_buckets[row] = (float)bucket;
        atomicAdd(&out_counts[bucket], 1.0f);
    }
}

extern "C" void kernel_launch(void* const* d_in, const int* in_sizes, int n_in,
                              void* d_out, int out_size, void* d_ws, size_t ws_size,
                              hipStream_t stream) {
    const float* items  = (const float*)d_in[0];   // [N, 64]
    const float* W      = (const float*)d_in[1];   // [64, 16]
    const float* counts = (const float*)d_in[2];   // [65536]

    const int n_items = in_sizes[0] / DIM;         // 4,194,304

    float* out_buckets = (float*)d_out;
    float* out_counts  = out_buckets + n_items;

    hc_init_counts<<<(NUM_BUCKETS + 255) / 256, 256, 0, stream>>>(counts, out_counts);

    // 16 rows per wave, 8 waves per 256-thread block -> 128 rows per block.
    const int rows_per_block = 128;
    const int blocks = (n_items + rows_per_block - 1) / rows_per_block;  // 32768
    hc_hash_count<<<blocks, 256, 0, stream>>>(items, W, out_buckets, out_counts, n_items);
}